// DirectedGinPredictorAgent_34256659153343
// MI455X (gfx1250) — compile-verified
//
#include <hip/hip_runtime.h>
#include <hip/hip_bf16.h>
#include <math.h>

// ---------------------------------------------------------------------------
// Directed-GIN predictor for MI455X (gfx1250, wave32).
// Edge aggregation via L2-resident f32 atomics; node MLPs via
// V_WMMA_F32_16X16X4_F32 with LDS-staged, transposed, zero-padded weights so
// the inner WMMA loop has zero divergent guards. BN stats fused.
// ---------------------------------------------------------------------------

typedef float v2f __attribute__((ext_vector_type(2)));
typedef float v8f __attribute__((ext_vector_type(8)));

#define NN    250000
#define EE    2500000
#define GG    512
#define HH    32
#define RS    34           // padded LDS row stride (words) -> conflict-free
#define WPB   8            // waves per block in the WMMA kernel

__device__ __forceinline__ v8f wmma_f32_k4(v2f a, v2f b, v8f c) {
  // (neg_a, A, neg_b, B, c_mod, C, reuse_a, reuse_b)
  return __builtin_amdgcn_wmma_f32_16x16x4_f32(false, a, false, b, (short)0, c,
                                               false, false);
}

// D(16x32) = In(16xK) * W(Kx32) + bias, as two 16x16 WMMA column tiles.
// tb: In tile in LDS, row-major stride RS, zero-padded K columns.
// wt: W^T in LDS: wt[col*RS + k], zero-padded k rows. No guards needed.
template <int KCH>
__device__ __forceinline__ void gemm_tile(const float* tb, const float* wt,
                                          const float* __restrict__ bias,
                                          int lane, v8f& acc0, v8f& acc1) {
  const int l16 = lane & 15;
  const int hs  = lane >> 4;
  const float bv0 = bias[l16];
  const float bv1 = bias[16 + l16];
#pragma unroll
  for (int j = 0; j < 8; ++j) { acc0[j] = bv0; acc1[j] = bv1; }
#pragma unroll
  for (int k = 0; k < KCH; ++k) {
    const int kk = 4 * k + 2 * hs;           // K pair handled by this lane
    v2f a  = *(const v2f*)&tb[l16 * RS + kk];
    v2f b0 = *(const v2f*)&wt[l16 * RS + kk];
    v2f b1 = *(const v2f*)&wt[(l16 + 16) * RS + kk];
    acc0 = wmma_f32_k4(a, b0, acc0);
    acc1 = wmma_f32_k4(a, b1, acc1);
  }
}

// ---------------------------------------------------------------------------
__global__ void zero_kernel(float* __restrict__ p, int n) {
  int i = blockIdx.x * blockDim.x + threadIdx.x;
  if (i < n) p[i] = 0.0f;
}

// Both directions in one pass: aggF[dst] += h[src], aggB[src] += h[dst].
__global__ void edge_scatter_kernel(const float* __restrict__ h,
                                    const int* __restrict__ src,
                                    const int* __restrict__ dst,
                                    float* __restrict__ aggF,
                                    float* __restrict__ aggB,
                                    int E, int d, int lpad) {
  unsigned tid = blockIdx.x * blockDim.x + threadIdx.x;
  int e = (int)(tid >> lpad);
  int c = (int)(tid & ((1u << lpad) - 1u));
  if (e >= E || c >= d) return;
  int s = src[e], t = dst[e];
  atomicAdd(&aggF[t * d + c], h[s * d + c]);
  atomicAdd(&aggB[s * d + c], h[t * d + c]);
}

// One wave per 16-node tile: o = 0.5*(relu(ginF)+relu(ginB)),
// ginX = relu((x+aggX)@W1+b1)@W2+b2 ; writes o and accumulates BN column stats.
template <int DIN, int KPAD, int KCH>
__global__ void __launch_bounds__(WPB * 32)
dgin_mlp_kernel(const float* __restrict__ xin,
                const float* __restrict__ aggF, const float* __restrict__ aggB,
                const float* __restrict__ WF1, const float* __restrict__ bF1,
                const float* __restrict__ WF2, const float* __restrict__ bF2,
                const float* __restrict__ WB1, const float* __restrict__ bB1,
                const float* __restrict__ WB2, const float* __restrict__ bB2,
                float* __restrict__ hpre,
                float* __restrict__ colsum, float* __restrict__ colsq,
                int nrows) {
  __shared__ float tiles[WPB * 16 * RS];     // per-wave In / h1 tiles
  __shared__ float wlds[4 * 32 * RS];        // W1F^T, W2F^T, W1B^T, W2B^T
  float* wt1f = &wlds[0 * 32 * RS];
  float* wt2f = &wlds[1 * 32 * RS];
  float* wt1b = &wlds[2 * 32 * RS];
  float* wt2b = &wlds[3 * 32 * RS];

  // Stage weights transposed + zero-padded to K=32 (one time per block).
  for (int idx = threadIdx.x; idx < 32 * 32; idx += WPB * 32) {
    int r = idx >> 5;          // K row
    int c = idx & 31;          // output col
    float w1f = 0.0f, w1b = 0.0f;
    if (r < DIN) { w1f = WF1[r * 32 + c]; w1b = WB1[r * 32 + c]; }
    wt1f[c * RS + r] = w1f;
    wt1b[c * RS + r] = w1b;
    wt2f[c * RS + r] = WF2[r * 32 + c];
    wt2b[c * RS + r] = WB2[r * 32 + c];
  }

  const int lane = threadIdx.x & 31;
  const int wave = threadIdx.x >> 5;
  const int ntiles = nrows >> 4;
  const int tileRaw = blockIdx.x * WPB + wave;
  const bool active = tileRaw < ntiles;         // wave-uniform
  const int tile = active ? tileRaw : (ntiles - 1);
  float* tb = &tiles[wave * 16 * RS];
  const int m0  = tile << 4;
  const int l16 = lane & 15;
  const int hs  = lane >> 4;

  __syncthreads();

  v8f ginF0 = {}, ginF1 = {}, ginB0 = {}, ginB1 = {};

#pragma unroll
  for (int dir = 0; dir < 2; ++dir) {
    const float* agg = dir ? aggB : aggF;
    const float* wt1 = dir ? wt1b : wt1f;
    const float* b1  = dir ? bB1 : bF1;
    const float* wt2 = dir ? wt2b : wt2f;
    const float* b2  = dir ? bB2 : bF2;

    // Stage In = x + agg (16 x KPAD, zero-padded) into this wave's LDS tile.
#pragma unroll
    for (int idx = lane; idx < 16 * KPAD; idx += 32) {
      int r = idx / KPAD;
      int c = idx - r * KPAD;
      float v = 0.0f;
      if (c < DIN) v = xin[(m0 + r) * DIN + c] + agg[(m0 + r) * DIN + c];
      tb[r * RS + c] = v;
    }
    __syncthreads();

    v8f a0, a1;
    gemm_tile<KCH>(tb, wt1, b1, lane, a0, a1);
#pragma unroll
    for (int j = 0; j < 8; ++j) {
      a0[j] = fmaxf(a0[j], 0.0f);
      a1[j] = fmaxf(a1[j], 0.0f);
    }
    __syncthreads();
    // h1 (C/D layout) -> LDS row-major, reuse the same tile.
#pragma unroll
    for (int j = 0; j < 8; ++j) {
      int r = j + 8 * hs;
      tb[r * RS + l16]      = a0[j];
      tb[r * RS + l16 + 16] = a1[j];
    }
    __syncthreads();

    v8f o0, o1;
    gemm_tile<8>(tb, wt2, b2, lane, o0, o1);
    if (dir == 0) { ginF0 = o0; ginF1 = o1; }
    else          { ginB0 = o0; ginB1 = o1; }
    __syncthreads();
  }

  v8f o0, o1;
#pragma unroll
  for (int j = 0; j < 8; ++j) {
    o0[j] = 0.5f * (fmaxf(ginF0[j], 0.0f) + fmaxf(ginB0[j], 0.0f));
    o1[j] = 0.5f * (fmaxf(ginF1[j], 0.0f) + fmaxf(ginB1[j], 0.0f));
  }

  // write pre-BN output
  if (active) {
#pragma unroll
    for (int j = 0; j < 8; ++j) {
      int node = m0 + j + 8 * hs;
      hpre[node * HH + l16]      = o0[j];
      hpre[node * HH + l16 + 16] = o1[j];
    }
  }

  // fused BN column statistics
  float s0 = 0.f, s1 = 0.f, q0 = 0.f, q1 = 0.f;
#pragma unroll
  for (int j = 0; j < 8; ++j) {
    s0 += o0[j]; q0 += o0[j] * o0[j];
    s1 += o1[j]; q1 += o1[j] * o1[j];
  }
  s0 += __shfl_xor(s0, 16, 32);
  s1 += __shfl_xor(s1, 16, 32);
  q0 += __shfl_xor(q0, 16, 32);
  q1 += __shfl_xor(q1, 16, 32);
  if (active && lane < 16) {
    atomicAdd(&colsum[lane],      s0);
    atomicAdd(&colsum[lane + 16], s1);
    atomicAdd(&colsq[lane],       q0);
    atomicAdd(&colsq[lane + 16],  q1);
  }
}

__global__ void bn_finalize_kernel(const float* __restrict__ colsum,
                                   const float* __restrict__ colsq,
                                   const float* __restrict__ g,
                                   const float* __restrict__ b,
                                   float* __restrict__ scale,
                                   float* __restrict__ shift, float invN) {
  int c = threadIdx.x;
  if (c < HH) {
    float mu  = colsum[c] * invN;
    float var = colsq[c] * invN - mu * mu;
    float sc  = g[c] * rsqrtf(var + 1e-5f);
    scale[c] = sc;
    shift[c] = b[c] - mu * sc;
  }
}

__global__ void bn_apply_kernel(float* __restrict__ h,
                                const float* __restrict__ scale,
                                const float* __restrict__ shift, int n) {
  int i = blockIdx.x * blockDim.x + threadIdx.x;
  if (i < n) {
    int c = i & (HH - 1);
    h[i] = h[i] * scale[c] + shift[c];
  }
}

__global__ void pool_scatter_kernel(const float* __restrict__ h,
                                    const int* __restrict__ batch,
                                    float* __restrict__ sums,
                                    float* __restrict__ cnts, int N) {
  unsigned tid = blockIdx.x * blockDim.x + threadIdx.x;
  int i = (int)(tid >> 5);
  int c = (int)(tid & 31u);
  if (i >= N) return;
  int g = batch[i];
  atomicAdd(&sums[g * HH + c], h[i * HH + c]);
  if (c == 0) atomicAdd(&cnts[g], 1.0f);
}

__global__ void head_kernel(const float* __restrict__ sums,
                            const float* __restrict__ cnts,
                            const float* __restrict__ lbW,
                            const float* __restrict__ lbb,
                            const float* __restrict__ lmW,
                            const float* __restrict__ lmb,
                            float* __restrict__ out, int G) {
  int g = blockIdx.x * blockDim.x + threadIdx.x;
  if (g >= G) return;
  float inv = 1.0f / fmaxf(cnts[g], 1.0f);
  float p[HH];
#pragma unroll
  for (int c = 0; c < HH; ++c) p[c] = sums[g * HH + c] * inv;
  float logit = lmb[0];
#pragma unroll
  for (int j = 0; j < 16; ++j) {
    float e = lbb[j];
#pragma unroll
    for (int c = 0; c < HH; ++c) e += p[c] * lbW[c * 16 + j];
    e = fmaxf(e, 0.0f);
    logit += e * lmW[j];
  }
  out[g] = 1.0f / (1.0f + expf(-logit));
}

// ---------------------------------------------------------------------------
extern "C" void kernel_launch(void* const* d_in, const int* in_sizes, int n_in,
                              void* d_out, int out_size, void* d_ws, size_t ws_size,
                              hipStream_t stream) {
  (void)in_sizes; (void)n_in; (void)out_size; (void)ws_size;
  const float* x     = (const float*)d_in[0];
  const int*   ei    = (const int*)d_in[1];
  const int*   batch = (const int*)d_in[2];
  const int* src = ei;
  const int* dst = ei + EE;

  float* ws = (float*)d_ws;
  const size_t NH = (size_t)NN * HH;          // 8,000,000
  float* aggF   = ws;
  float* aggB   = ws + NH;
  float* buf0   = ws + 2 * NH;
  float* buf1   = ws + 3 * NH;
  float* colsum = ws + 4 * NH;                // 32
  float* colsq  = colsum + 32;                // 32
  float* scale  = colsum + 64;                // 32
  float* shift  = colsum + 96;                // 32
  float* psums  = colsum + 128;               // G*32
  float* pcnts  = psums + GG * HH;            // G

  const int TPB = 256;
  auto cdiv = [](long long a, long long b) { return (int)((a + b - 1) / b); };

  const float* layer_in[3] = { x, buf0, buf1 };
  float*       layer_out[3] = { buf0, buf1, buf0 };
  const int din_l[3]  = { 6, HH, HH };
  const int lpad_l[3] = { 3, 5, 5 };
  const int ntiles = NN / 16;
  const int mlp_blocks = cdiv(ntiles, WPB);

  for (int l = 0; l < 3; ++l) {
    const int base = 3 + l * 10;
    const float* WF1 = (const float*)d_in[base + 0];
    const float* bF1 = (const float*)d_in[base + 1];
    const float* WF2 = (const float*)d_in[base + 2];
    const float* bF2 = (const float*)d_in[base + 3];
    const float* WB1 = (const float*)d_in[base + 4];
    const float* bB1 = (const float*)d_in[base + 5];
    const float* WB2 = (const float*)d_in[base + 6];
    const float* bB2 = (const float*)d_in[base + 7];
    const float* bng = (const float*)d_in[base + 8];
    const float* bnb = (const float*)d_in[base + 9];

    const int din = din_l[l];
    const int nz = 2 * NN * din;  // aggF + aggB contiguous
    zero_kernel<<<cdiv(nz, TPB), TPB, 0, stream>>>(aggF, nz);
    zero_kernel<<<1, 64, 0, stream>>>(colsum, 64);

    long long sthreads = (long long)EE << lpad_l[l];
    edge_scatter_kernel<<<cdiv(sthreads, TPB), TPB, 0, stream>>>(
        layer_in[l], src, dst, aggF, aggB, EE, din, lpad_l[l]);

    if (l == 0) {
      dgin_mlp_kernel<6, 8, 2><<<mlp_blocks, WPB * 32, 0, stream>>>(
          layer_in[l], aggF, aggB,
          WF1, bF1, WF2, bF2, WB1, bB1, WB2, bB2,
          layer_out[l], colsum, colsq, NN);
    } else {
      dgin_mlp_kernel<32, 32, 8><<<mlp_blocks, WPB * 32, 0, stream>>>(
          layer_in[l], aggF, aggB,
          WF1, bF1, WF2, bF2, WB1, bB1, WB2, bB2,
          layer_out[l], colsum, colsq, NN);
    }

    bn_finalize_kernel<<<1, 32, 0, stream>>>(colsum, colsq, bng, bnb, scale,
                                             shift, 1.0f / (float)NN);
    bn_apply_kernel<<<cdiv((long long)NN * HH, TPB), TPB, 0, stream>>>(
        layer_out[l], scale, shift, NN * HH);
  }

  // global mean pool + head
  zero_kernel<<<cdiv(GG * HH + GG, TPB), TPB, 0, stream>>>(psums, GG * HH + GG);
  pool_scatter_kernel<<<cdiv((long long)NN * HH, TPB), TPB, 0, stream>>>(
      buf0, batch, psums, pcnts, NN);
  head_kernel<<<cdiv(GG, TPB), TPB, 0, stream>>>(
      psums, pcnts,
      (const float*)d_in[33], (const float*)d_in[34],
      (const float*)d_in[35], (const float*)d_in[36],
      (float*)d_out, GG);
}